// QuadrupletLoss_80161269612716
// MI455X (gfx1250) — compile-verified
//
#include <hip/hip_runtime.h>
#include <hip/hip_bf16.h>
#include <math.h>

typedef __attribute__((ext_vector_type(2))) float v2f;
typedef __attribute__((ext_vector_type(8))) float v8f;

#define NPART 64          // partner-column panel width for the WMMA GEMM
#define MARGIN1 0.3f
#define MARGIN2 0.3f
#define MAXQ 10

// ---------------- helpers ----------------

__device__ __forceinline__ float wave_reduce_add(float s) {
#pragma unroll
  for (int off = 16; off > 0; off >>= 1) s += __shfl_xor(s, off, 32);
  return s;
}

// Wave-parallel dot of two d-length fp32 rows (d % 128 == 0). All 32 lanes
// of the wave call this together; result valid in all lanes.
__device__ __forceinline__ float wave_dot(const float* __restrict__ a,
                                          const float* __restrict__ b,
                                          int lane, int d) {
  float s = 0.f;
  for (int k = lane * 4; k < d; k += 128) {
    const float4 va = *(const float4*)(a + k);
    const float4 vb = *(const float4*)(b + k);
    s = fmaf(va.x, vb.x, s);
    s = fmaf(va.y, vb.y, s);
    s = fmaf(va.z, vb.z, s);
    s = fmaf(va.w, vb.w, s);
  }
  return wave_reduce_add(s);
}

// ---------------- kernel 1: row squared norms ----------------

__global__ void row_sq_kernel(const float* __restrict__ E, float* __restrict__ sq,
                              int n, int d) {
  const int gwave = (int)((blockIdx.x * blockDim.x + threadIdx.x) >> 5);
  const int lane  = threadIdx.x & 31;
  if (gwave >= n) return;
  const float* row = E + (size_t)gwave * d;
  float s = 0.f;
  for (int k = lane * 4; k < d; k += 128) {
    const float4 v = *(const float4*)(row + k);
    s = fmaf(v.x, v.x, s);
    s = fmaf(v.y, v.y, s);
    s = fmaf(v.z, v.z, s);
    s = fmaf(v.w, v.w, s);
  }
  s = wave_reduce_add(s);
  if (lane == 0) sq[gwave] = s;
}

// ---------------- kernel 2: label-only index selection ----------------
// For each anchor i: first same-label j != i (pos), first MAXQ+1 different-
// label j (negatives). Mirrors argmax/top_k-of-indices in the reference;
// missing entries fall back to n-1 (JAX clamps OOB gathers).

__global__ void index_kernel(const int* __restrict__ labels, int n,
                             int* __restrict__ posIdx, int* __restrict__ hasPos,
                             int* __restrict__ negs) {
  const int i = blockIdx.x * blockDim.x + threadIdx.x;
  if (i >= n) return;
  const int li = labels[i];
  int p = -1;
  int cnt = 0;
  int nn[MAXQ + 1];
  for (int j = 0; j < n; ++j) {
    if (p >= 0 && cnt >= MAXQ + 1) break;
    const int lj = labels[j];
    if (lj == li) {
      if (j != i && p < 0) p = j;
    } else if (cnt < MAXQ + 1) {
      nn[cnt++] = j;
    }
  }
  hasPos[i] = (p >= 0) ? 1 : 0;
  posIdx[i] = (p >= 0) ? p : 0;
  for (int k = cnt; k < MAXQ + 1; ++k) nn[k] = n - 1;
#pragma unroll
  for (int k = 0; k < MAXQ + 1; ++k) negs[(size_t)i * (MAXQ + 1) + k] = nn[k];
}

// ---------------- kernel 3: skinny FP32 WMMA GEMM ----------------
// P[i][j] = dot(E[i], E[j]) for i in [0,n), j in [0,NPART).
// One block (128 thr / 4 waves) per 16-row anchor tile; K is split across
// the 4 waves; each wave accumulates 4 C tiles (partner cols 0..63) with
// V_WMMA_F32_16X16X4_F32; partial C tiles are reduced through LDS.
//
// FP32 A 16x4 fragment (per ISA): lanes 0-15 hold row M=lane, (K=k, k+1);
// lanes 16-31 hold row M=lane-16, (K=k+2, k+3). B mirrors with N<->M, so
// both fragments are a float2 load at row_base + k + (hi ? 2 : 0).
//
// The K-loop body is shared; KC is the per-wave K chunk as a compile-time
// constant (0 => runtime fallback). Constant trip count lets the 24-bit
// global IOFFSET absorb the per-step offsets (no v_add_u64 per iteration)
// and enables real unrolling (32 static WMMAs / branch).

template <int KC>
__device__ __forceinline__ void gram_body(
    const float* __restrict__ E, float* __restrict__ P, int d, int kChunkRt) {
  __shared__ float lds[4][4][256];

  const int wave = threadIdx.x >> 5;
  const int lane = threadIdx.x & 31;
  const int lsub = lane & 15;
  const bool hi  = lane >= 16;
  const int rowBase = blockIdx.x * 16;

  const int kChunk = (KC > 0) ? KC : kChunkRt;   // K split across 4 waves
  const int k0 = wave * kChunk;

  const float* arow = E + (size_t)(rowBase + lsub) * d + k0 + (hi ? 2 : 0);
  const float* b0 = E + (size_t)(0  + lsub) * d + k0 + (hi ? 2 : 0);
  const float* b1 = E + (size_t)(16 + lsub) * d + k0 + (hi ? 2 : 0);
  const float* b2 = E + (size_t)(32 + lsub) * d + k0 + (hi ? 2 : 0);
  const float* b3 = E + (size_t)(48 + lsub) * d + k0 + (hi ? 2 : 0);

  v8f c0 = {}, c1 = {}, c2 = {}, c3 = {};
#pragma unroll 8
  for (int k = 0; k < kChunk; k += 4) {
    const v2f a  = *(const v2f*)(arow + k);
    const v2f x0 = *(const v2f*)(b0 + k);
    const v2f x1 = *(const v2f*)(b1 + k);
    const v2f x2 = *(const v2f*)(b2 + k);
    const v2f x3 = *(const v2f*)(b3 + k);
    c0 = __builtin_amdgcn_wmma_f32_16x16x4_f32(false, a, false, x0, (short)0, c0, false, false);
    c1 = __builtin_amdgcn_wmma_f32_16x16x4_f32(false, a, false, x1, (short)0, c1, false, false);
    c2 = __builtin_amdgcn_wmma_f32_16x16x4_f32(false, a, false, x2, (short)0, c2, false, false);
    c3 = __builtin_amdgcn_wmma_f32_16x16x4_f32(false, a, false, x3, (short)0, c3, false, false);
  }

#pragma unroll
  for (int r = 0; r < 8; ++r) {
    lds[wave][0][r * 32 + lane] = c0[r];
    lds[wave][1][r * 32 + lane] = c1[r];
    lds[wave][2][r * 32 + lane] = c2[r];
    lds[wave][3][r * 32 + lane] = c3[r];
  }
  __syncthreads();

  // C/D 16x16 f32 layout: VGPR r -> M=r (lanes 0-15) / M=8+r (lanes 16-31),
  // N = lane & 15. Sum the 4 K-partials and scatter to P (row-major n x 64).
  for (int idx = threadIdx.x; idx < 4 * 256; idx += 128) {
    const int t = idx >> 8;        // partner tile 0..3
    const int e = idx & 255;       // element within tile
    const int r = e >> 5;
    const int l = e & 31;
    const int m = ((l >= 16) ? 8 : 0) + r;
    const int ncol = t * 16 + (l & 15);
    const float s = lds[0][t][e] + lds[1][t][e] + lds[2][t][e] + lds[3][t][e];
    P[(size_t)(rowBase + m) * NPART + ncol] = s;
  }
}

// Specialized for d=2048 (kChunk = 512 as a constant).
__global__ __launch_bounds__(128) void gram_partner_wmma_2048(
    const float* __restrict__ E, float* __restrict__ P) {
  gram_body<512>(E, P, 2048, 0);
}

// Runtime-shape fallback (d % 16 == 0).
__global__ __launch_bounds__(128) void gram_partner_wmma_rt(
    const float* __restrict__ E, float* __restrict__ P, int d) {
  gram_body<0>(E, P, d, d >> 2);
}

// ---------------- kernel 4: per-anchor quadruplet loss ----------------
// One wave per anchor. d_ap via wave_dot (arbitrary partner); d_an and
// d(n0, nk) come from P when the index is inside the 64-wide panel
// (always, in practice), else exact wave_dot fallback.

__global__ void loss_kernel(const float* __restrict__ E, const float* __restrict__ P,
                            const float* __restrict__ sq,
                            const int* __restrict__ posIdx,
                            const int* __restrict__ hasPos,
                            const int* __restrict__ negs,
                            float* __restrict__ perAnchor, int n, int d) {
  const int i   = (int)((blockIdx.x * blockDim.x + threadIdx.x) >> 5);
  const int lane = threadIdx.x & 31;
  if (i >= n) return;
  if (!hasPos[i]) {
    if (lane == 0) perAnchor[i] = 0.f;
    return;
  }

  const int p = posIdx[i];
  const float dap = wave_dot(E + (size_t)i * d, E + (size_t)p * d, lane, d);
  const float d_ap = sqrtf(fmaxf(sq[i] + sq[p] - 2.f * dap, 0.f));

  int n0 = negs[(size_t)i * (MAXQ + 1)];
  if (n0 >= n) n0 = n - 1;
  const float dan = (n0 < NPART)
      ? P[(size_t)i * NPART + n0]
      : wave_dot(E + (size_t)i * d, E + (size_t)n0 * d, lane, d);
  const float d_an = sqrtf(fmaxf(sq[i] + sq[n0] - 2.f * dan, 0.f));

  float acc = (float)MAXQ * fmaxf(d_ap - d_an + MARGIN1, 0.f);
#pragma unroll
  for (int q = 1; q <= MAXQ; ++q) {
    int nk = negs[(size_t)i * (MAXQ + 1) + q];
    if (nk >= n) nk = n - 1;
    const float dnn = (nk < NPART)
        ? P[(size_t)n0 * NPART + nk]
        : wave_dot(E + (size_t)n0 * d, E + (size_t)nk * d, lane, d);
    const float d12 = sqrtf(fmaxf(sq[n0] + sq[nk] - 2.f * dnn, 0.f));
    acc += fmaxf(d_ap - d12 + MARGIN2, 0.f);
  }
  if (lane == 0) perAnchor[i] = acc;
}

// ---------------- kernel 5: deterministic final reduction ----------------

__global__ void final_reduce(const float* __restrict__ perAnchor,
                             const int* __restrict__ hasPos,
                             float* __restrict__ out, int n) {
  __shared__ float ssum[256];
  __shared__ int   scnt[256];
  const int t = threadIdx.x;
  float s = 0.f;
  int c = 0;
  for (int i = t; i < n; i += 256) {
    s += perAnchor[i];
    c += hasPos[i];
  }
  ssum[t] = s;
  scnt[t] = c;
  __syncthreads();
#pragma unroll
  for (int off = 128; off > 0; off >>= 1) {
    if (t < off) {
      ssum[t] += ssum[t + off];
      scnt[t] += scnt[t + off];
    }
    __syncthreads();
  }
  if (t == 0) {
    const long long count = (long long)scnt[0] * MAXQ;
    out[0] = (count > 0) ? (ssum[0] / (float)count) : 0.f;
  }
}

// ---------------- launch ----------------

extern "C" void kernel_launch(void* const* d_in, const int* in_sizes, int n_in,
                              void* d_out, int out_size, void* d_ws, size_t ws_size,
                              hipStream_t stream) {
  (void)n_in; (void)out_size; (void)ws_size;
  const float* E      = (const float*)d_in[0];
  const int*   labels = (const int*)d_in[1];
  const int n = in_sizes[1];
  const int d = in_sizes[0] / n;   // 4096 x 2048

  // Workspace layout (~1.3 MB for n=4096).
  char* ws = (char*)d_ws;
  size_t off = 0;
  float* sq      = (float*)(ws + off); off += (size_t)n * sizeof(float);
  int*   posIdx  = (int*)(ws + off);   off += (size_t)n * sizeof(int);
  int*   hasPos  = (int*)(ws + off);   off += (size_t)n * sizeof(int);
  int*   negs    = (int*)(ws + off);   off += (size_t)n * (MAXQ + 1) * sizeof(int);
  off = (off + 255) & ~(size_t)255;
  float* P       = (float*)(ws + off); off += (size_t)n * NPART * sizeof(float);
  float* perAnch = (float*)(ws + off); off += (size_t)n * sizeof(float);

  const int wavesPerBlock = 8;                       // 256 threads
  const int waveBlocks = (n + wavesPerBlock - 1) / wavesPerBlock;

  row_sq_kernel<<<waveBlocks, 256, 0, stream>>>(E, sq, n, d);
  index_kernel<<<(n + 255) / 256, 256, 0, stream>>>(labels, n, posIdx, hasPos, negs);
  if (d == 2048) {
    gram_partner_wmma_2048<<<n / 16, 128, 0, stream>>>(E, P);
  } else {
    gram_partner_wmma_rt<<<n / 16, 128, 0, stream>>>(E, P, d);
  }
  loss_kernel<<<waveBlocks, 256, 0, stream>>>(E, P, sq, posIdx, hasPos, negs, perAnch, n, d);
  final_reduce<<<1, 256, 0, stream>>>(perAnch, hasPos, (float*)d_out, n);
}